// GraphConvolution_G_85916525789263
// MI455X (gfx1250) — compile-verified
//
#include <hip/hip_runtime.h>
#include <cmath>

typedef __attribute__((ext_vector_type(2))) float v2f;
typedef __attribute__((ext_vector_type(8))) float v8f;

#define FDIM 256   // feature dim (in == out)

// ---------------------------------------------------------------------------
// Kernel 1: zero the hi accumulator (lives in d_out)
// ---------------------------------------------------------------------------
__global__ __launch_bounds__(256) void zero_f4(float4* __restrict__ p, int n4) {
    int i = blockIdx.x * 256 + threadIdx.x;
    if (i < n4) p[i] = make_float4(0.f, 0.f, 0.f, 0.f);
}

// ---------------------------------------------------------------------------
// Kernel 2: COO SpMM scatter: hi[row,:] += val * x[col,:]
// One wave32 per edge; each lane moves 8 floats (2 x float4 gather,
// 8 native f32 atomic adds). x rows are mostly L2-resident (102MB vs 192MB L2).
// ---------------------------------------------------------------------------
__global__ __launch_bounds__(256) void spmm_scatter(
    const float* __restrict__ x, const int* __restrict__ rows,
    const int* __restrict__ cols, const float* __restrict__ vals,
    float* __restrict__ hi, int E)
{
    int edge = blockIdx.x * 8 + (threadIdx.x >> 5);
    int lane = threadIdx.x & 31;
    if (edge >= E) return;
    int   r = rows[edge];
    int   c = cols[edge];
    float v = vals[edge];
    const float4* xs = (const float4*)(x + (size_t)c * FDIM);
    float* dst = hi + (size_t)r * FDIM;
    float4 x0 = xs[lane];
    float4 x1 = xs[lane + 32];
    int b0 = lane * 4;
    int b1 = 128 + lane * 4;
    unsafeAtomicAdd(dst + b0 + 0, v * x0.x);
    unsafeAtomicAdd(dst + b0 + 1, v * x0.y);
    unsafeAtomicAdd(dst + b0 + 2, v * x0.z);
    unsafeAtomicAdd(dst + b0 + 3, v * x0.w);
    unsafeAtomicAdd(dst + b1 + 0, v * x1.x);
    unsafeAtomicAdd(dst + b1 + 1, v * x1.y);
    unsafeAtomicAdd(dst + b1 + 2, v * x1.z);
    unsafeAtomicAdd(dst + b1 + 3, v * x1.w);
}

// ---------------------------------------------------------------------------
// Kernel 3: pre-swizzle W (256x256) into WMMA B-fragment layout in d_ws.
// Layout: Wp[ ((kb*2+half)*32 + lane)*16 + ntLocal*2 + {0,1} ]
//   = W[4*kb + 2*(lane>>4) + {0,1}][half*128 + ntLocal*16 + (lane&15)]
// so in the GEMM each lane reads its 8 N-tile fragments for a K-step as
// 64B contiguous (4 x global_load_b128).
// ---------------------------------------------------------------------------
__global__ __launch_bounds__(256) void pack_w(const float* __restrict__ W,
                                              float* __restrict__ Wp)
{
    int t = blockIdx.x * 256 + threadIdx.x;        // 0 .. 32767
    int ntL  = t & 7;
    int lane = (t >> 3) & 31;
    int half = (t >> 8) & 1;
    int kb   = t >> 9;                             // 0 .. 63
    int col  = half * 128 + ntL * 16 + (lane & 15);
    int krow = kb * 4 + 2 * (lane >> 4);
    Wp[2 * t + 0] = W[(size_t)krow * FDIM + col];
    Wp[2 * t + 1] = W[(size_t)(krow + 1) * FDIM + col];
}

// ---------------------------------------------------------------------------
// Kernel 4: fused GEMM + epilogue using V_WMMA_F32_16X16X4_F32.
// Block = 256 threads (8 waves), M-tile = 64 rows.
// support = (1-alpha)*hi + alpha*h0 is staged into LDS in A-fragment-packed
// layout (64KB, conflict-free ds_load_b64 in the K-loop).
// Wave w: row strip (w>>1)*16, column half (w&1)*128 -> 8 accumulator tiles.
// out = theta * acc + (1-theta) * support ; hi aliases out (panel staged
// through LDS before being overwritten).
// ---------------------------------------------------------------------------
__global__ __launch_bounds__(256) void gemm_fused(
    const float* hi, const float* __restrict__ h0,
    const float* __restrict__ Wp, float* out,
    const float* __restrict__ lamda_p, const float* __restrict__ alpha_p,
    const int* __restrict__ l_p, int N)
{
    __shared__ float As[16384];   // 64 rows x 256 cols, A-fragment packed (64KB)

    int tid     = threadIdx.x;
    int rowBase = blockIdx.x * 64;
    float alpha = *alpha_p;
    float theta = logf(*lamda_p / (float)(*l_p) + 1.0f);
    float oma   = 1.0f - alpha;
    float omt   = 1.0f - theta;

    // ---- stage support panel into LDS (packed A-fragment layout) ----------
    // element A[r][c] lives at ((s*64 + c/4)*32 + (r&15) + 16*((c>>1)&1))*2 + (c&1)
    for (int i = tid; i < 64 * 64; i += 256) {
        int r  = i >> 6;        // local row 0..63
        int c4 = i & 63;        // float4 column index == kb
        int row = rowBase + r;
        float4 hv = make_float4(0.f, 0.f, 0.f, 0.f);
        float4 gv = make_float4(0.f, 0.f, 0.f, 0.f);
        if (row < N) {
            hv = ((const float4*)(hi + (size_t)row * FDIM))[c4];
            gv = ((const float4*)(h0 + (size_t)row * FDIM))[c4];
        }
        float4 sv;
        sv.x = oma * hv.x + alpha * gv.x;
        sv.y = oma * hv.y + alpha * gv.y;
        sv.z = oma * hv.z + alpha * gv.z;
        sv.w = oma * hv.w + alpha * gv.w;
        int s    = r >> 4;
        int rm   = r & 15;
        int base = ((s * 64 + c4) * 32);
        *(v2f*)&As[(base + rm) * 2]      = v2f{sv.x, sv.y};   // khalf = 0
        *(v2f*)&As[(base + rm + 16) * 2] = v2f{sv.z, sv.w};   // khalf = 1
    }
    __syncthreads();

    int wave  = tid >> 5;
    int lane  = tid & 31;
    int m     = lane & 15;
    int khalf = lane >> 4;
    int sIdx  = wave >> 1;       // row strip 0..3
    int half  = wave & 1;        // column half

    v8f acc[8] = {};

    const v2f* Ap = (const v2f*)As;

    // ---- K loop: 64 steps of K=4, 8 WMMAs per step ------------------------
    for (int kb = 0; kb < 64; ++kb) {
        v2f a = Ap[(sIdx * 64 + kb) * 32 + lane];
        const float4* fb = (const float4*)(Wp + (size_t)(((kb * 2 + half) * 32 + lane) * 16));
        float4 q0 = fb[0], q1 = fb[1], q2 = fb[2], q3 = fb[3];
        v2f b0 = v2f{q0.x, q0.y}, b1 = v2f{q0.z, q0.w};
        v2f b2 = v2f{q1.x, q1.y}, b3 = v2f{q1.z, q1.w};
        v2f b4 = v2f{q2.x, q2.y}, b5 = v2f{q2.z, q2.w};
        v2f b6 = v2f{q3.x, q3.y}, b7 = v2f{q3.z, q3.w};
        acc[0] = __builtin_amdgcn_wmma_f32_16x16x4_f32(false, a, false, b0, (short)0, acc[0], false, false);
        acc[1] = __builtin_amdgcn_wmma_f32_16x16x4_f32(false, a, false, b1, (short)0, acc[1], false, false);
        acc[2] = __builtin_amdgcn_wmma_f32_16x16x4_f32(false, a, false, b2, (short)0, acc[2], false, false);
        acc[3] = __builtin_amdgcn_wmma_f32_16x16x4_f32(false, a, false, b3, (short)0, acc[3], false, false);
        acc[4] = __builtin_amdgcn_wmma_f32_16x16x4_f32(false, a, false, b4, (short)0, acc[4], false, false);
        acc[5] = __builtin_amdgcn_wmma_f32_16x16x4_f32(false, a, false, b5, (short)0, acc[5], false, false);
        acc[6] = __builtin_amdgcn_wmma_f32_16x16x4_f32(false, a, false, b6, (short)0, acc[6], false, false);
        acc[7] = __builtin_amdgcn_wmma_f32_16x16x4_f32(false, a, false, b7, (short)0, acc[7], false, false);
    }

    // ---- epilogue: out = theta*acc + (1-theta)*support --------------------
    #pragma unroll
    for (int t = 0; t < 8; ++t) {
        int col = half * 128 + t * 16 + m;
        #pragma unroll
        for (int v = 0; v < 8; ++v) {
            int rloc = sIdx * 16 + v + 8 * khalf;
            int row  = rowBase + rloc;
            if (row < N) {
                float sres = As[((sIdx * 64 + (col >> 2)) * 32 +
                                 (rloc & 15) + 16 * ((col >> 1) & 1)) * 2 + (col & 1)];
                out[(size_t)row * FDIM + col] = theta * acc[t][v] + omt * sres;
            }
        }
    }
}

// ---------------------------------------------------------------------------
extern "C" void kernel_launch(void* const* d_in, const int* in_sizes, int n_in,
                              void* d_out, int out_size, void* d_ws, size_t ws_size,
                              hipStream_t stream) {
    const float* x        = (const float*)d_in[0];
    const int*   adj_rows = (const int*)d_in[1];
    const int*   adj_cols = (const int*)d_in[2];
    const float* adj_vals = (const float*)d_in[3];
    const float* h0       = (const float*)d_in[4];
    const float* weight   = (const float*)d_in[5];
    const float* lamda    = (const float*)d_in[6];
    const float* alpha    = (const float*)d_in[7];
    const int*   l        = (const int*)d_in[8];

    int N = in_sizes[0] / FDIM;     // 100000
    int E = in_sizes[3];            // 3200000

    float* hi = (float*)d_out;      // hi accumulates in d_out, then fused GEMM
    float* Wp = (float*)d_ws;       // 64KB*4 = 256KB packed W fragments

    // 1) zero hi accumulator
    int n4 = (N * FDIM) / 4;
    zero_f4<<<(n4 + 255) / 256, 256, 0, stream>>>((float4*)hi, n4);

    // 2) edge scatter (8 edges / 256-thread block, one wave32 per edge)
    spmm_scatter<<<(E + 7) / 8, 256, 0, stream>>>(x, adj_rows, adj_cols,
                                                  adj_vals, hi, E);

    // 3) pack W into B-fragment layout (32768 threads)
    pack_w<<<128, 256, 0, stream>>>(weight, Wp);

    // 4) fused blend + GEMM + residual epilogue
    gemm_fused<<<(N + 63) / 64, 256, 0, stream>>>(hi, h0, Wp, (float*)d_out,
                                                  lamda, alpha, l, N);
}